// ODEfunc_5257039970521
// MI455X (gfx1250) — compile-verified
//
#include <hip/hip_runtime.h>
#include <math.h>

typedef __attribute__((ext_vector_type(2))) float v2f;
typedef __attribute__((ext_vector_type(8))) float v8f;

#define HP 68  // padded row stride for per-wave activation tile (A-side reads)

// Native CDNA5 V_TANH_F32 (transcendental VALU) when available.
#if defined(__has_builtin)
#if __has_builtin(__builtin_amdgcn_tanhf)
#define FAST_TANH(x) __builtin_amdgcn_tanhf(x)
#elif __has_builtin(__builtin_amdgcn_tanh_f32)
#define FAST_TANH(x) __builtin_amdgcn_tanh_f32(x)
#endif
#endif
#ifndef FAST_TANH
#define FAST_TANH(x) tanhf(x)
#endif

static __device__ __forceinline__ v8f wmma4(v2f a, v2f b, v8f c) {
  // V_WMMA_F32_16X16X4_F32 : D = A(16x4,f32) x B(4x16,f32) + C(16x16,f32)
  return __builtin_amdgcn_wmma_f32_16x16x4_f32(
      /*neg_a=*/false, a, /*neg_b=*/false, b,
      /*c_mod=*/(short)0, c, /*reuse_a=*/false, /*reuse_b=*/false);
}

static __device__ __forceinline__ v2f lds_b64(const float* p) {
  const float2 v = *reinterpret_cast<const float2*>(p);
  v2f r; r.x = v.x; r.y = v.y;
  return r;
}

// Pair-interleaved B-operand layouts: element (k, col) of a KxN weight lives at
// [(k>>1)*2N + col*2 + (k&1)], so each lane's {row k, row k+1} pair at one col
// is a single aligned 8-byte LDS load straight into a WMMA B register pair.
__global__ __launch_bounds__(128) void odefunc_cnf_kernel(
    const float* __restrict__ tptr, const float* __restrict__ y,
    const float* __restrict__ W1, const float* __restrict__ b1,
    const float* __restrict__ W2, const float* __restrict__ b2,
    const float* __restrict__ W3, const float* __restrict__ b3,
    float* __restrict__ dy, float* __restrict__ negdiv) {
  __shared__ float sW1[6 * 128];    // z->h1, K padded 9->12 (6 row-pairs x 64 cols x 2)
  __shared__ float sW2[32 * 128];   // h1->h2 (32 row-pairs x 64 cols x 2)
  __shared__ float sG [32 * 128];   // G[i,k] = W2[i,k]*(W3@W1y)[k,i], K-dim = i
  __shared__ float sW3[32 * 32];    // h2->dy, N padded 8->16 (32 row-pairs x 16 x 2)
  __shared__ float sB1[64], sB2[64], sB3[8];
  __shared__ float sH[4][16 * HP];  // per-wave 16x64 activation staging

  const int tid = threadIdx.x;

  // ---- stage weights into LDS (zero the padded regions first) ----
  for (int i = tid; i < 6 * 128; i += 128) sW1[i] = 0.f;
  for (int i = tid; i < 32 * 32; i += 128) sW3[i] = 0.f;
  __syncthreads();
  for (int i = tid; i < 9 * 64; i += 128) {
    int r = i >> 6, c = i & 63;
    sW1[(r >> 1) * 128 + c * 2 + (r & 1)] = W1[i];
  }
  for (int i = tid; i < 64 * 64; i += 128) {
    int r = i >> 6, c = i & 63;
    sW2[(r >> 1) * 128 + c * 2 + (r & 1)] = W2[i];
  }
  for (int i = tid; i < 64 * 8; i += 128) {
    int k = i >> 3, d = i & 7;
    sW3[(k >> 1) * 32 + d * 2 + (k & 1)] = W3[i];
  }
  for (int i = tid; i < 64; i += 128) { sB1[i] = b1[i]; sB2[i] = b2[i]; }
  if (tid < 8) sB3[tid] = b3[tid];
  // G[i,k] = W2[i,k] * sum_d W3[k,d] * W1[1+d, i]   (trace(J) = a1^T G a2)
  for (int idx = tid; idx < 64 * 64; idx += 128) {
    int i = idx >> 6, k = idx & 63;
    float m = 0.f;
#pragma unroll
    for (int d = 0; d < 8; ++d) m += W3[k * 8 + d] * W1[(1 + d) * 64 + i];
    sG[(i >> 1) * 128 + k * 2 + (i & 1)] = W2[i * 64 + k] * m;
  }
  __syncthreads();

  const float tval = tptr[0];
  const int lane = tid & 31;
  const int wv   = tid >> 5;
  const int l15  = lane & 15;
  const int hi   = lane >> 4;   // 0: K=0,1 / rows 0-7 ; 1: K=2,3 / rows 8-15
  float* myH = sH[wv];
  const float4* __restrict__ y4 = reinterpret_cast<const float4*>(y);

  for (int tile = 0; tile < 4; ++tile) {
    const int base = blockIdx.x * 256 + wv * 64 + tile * 16;
    const int s = base + l15;

    // Branch-free z row: two coalesced 16B loads, then v_cndmask selects.
    const float4 ylo = y4[s * 2];
    const float4 yhi4 = y4[s * 2 + 1];
    // z = [t, y0..y7, 0,0,0]
    v2f a0, a1, a2;
    a0.x = hi ? ylo.y  : tval;    a0.y = hi ? ylo.z  : ylo.x;   // k {0,1}|{2,3}
    a1.x = hi ? yhi4.x : ylo.w;   a1.y = hi ? yhi4.y : yhi4.x;  // k {4,5}|{6,7} -> z4..z7
    // careful: z4=y3(ylo.w), z5=y4(yhi.x), z6=y5(yhi.y), z7=y6(yhi.z)
    a1.x = hi ? yhi4.y : ylo.w;   a1.y = hi ? yhi4.z : yhi4.x;
    a2.x = hi ? 0.f    : yhi4.w;  a2.y = 0.f;                   // k {8,9}|{10,11}

    // ---- layer 1: z @ W1 (K = 12 zero-padded) ----
    v8f acc1[4] = {};
    v2f aA[3] = {a0, a1, a2};
#pragma unroll
    for (int ks = 0; ks < 3; ++ks) {
      const int pr = ks * 2 + hi;
#pragma unroll
      for (int n = 0; n < 4; ++n) {
        const int col = n * 16 + l15;
        acc1[n] = wmma4(aA[ks], lds_b64(&sW1[pr * 128 + col * 2]), acc1[n]);
      }
    }
    // bias + tanh -> h1 staged to LDS
#pragma unroll
    for (int n = 0; n < 4; ++n) {
      const int col = n * 16 + l15;
      const float bb = sB1[col];
#pragma unroll
      for (int r = 0; r < 8; ++r)
        myH[(hi * 8 + r) * HP + col] = FAST_TANH(acc1[n][r] + bb);
    }
    __syncthreads();

    // ---- fused: h1 @ W2  and  v = a1 @ G  (a1 = 1 - h1^2 on load) ----
    v8f acc2[4] = {};
    v8f vacc[4] = {};
    for (int ks = 0; ks < 16; ++ks) {
      const int c0 = ks * 4 + 2 * hi;
      const int pr = ks * 2 + hi;
      const v2f ah = lds_b64(&myH[l15 * HP + c0]);
      v2f aa; aa.x = 1.f - ah.x * ah.x; aa.y = 1.f - ah.y * ah.y;
#pragma unroll
      for (int n = 0; n < 4; ++n) {
        const int col = n * 16 + l15;
        acc2[n] = wmma4(ah, lds_b64(&sW2[pr * 128 + col * 2]), acc2[n]);
        vacc[n] = wmma4(aa, lds_b64(&sG [pr * 128 + col * 2]), vacc[n]);
      }
    }
    __syncthreads();

    // ---- bias + tanh -> h2 ; divergence partials p = sum_k v[k]*(1-h2[k]^2) ----
    float p[8] = {0.f, 0.f, 0.f, 0.f, 0.f, 0.f, 0.f, 0.f};
#pragma unroll
    for (int n = 0; n < 4; ++n) {
      const int col = n * 16 + l15;
      const float bb = sB2[col];
#pragma unroll
      for (int r = 0; r < 8; ++r) {
        const float h = FAST_TANH(acc2[n][r] + bb);
        p[r] += vacc[n][r] * (1.f - h * h);
        myH[(hi * 8 + r) * HP + col] = h;   // overwrite h1 staging with h2
      }
    }
    __syncthreads();

    // ---- layer 3: h2 @ W3pad (N cols 8..15 zero) ----
    v8f acc3 = {};
    for (int ks = 0; ks < 16; ++ks) {
      const int c0 = ks * 4 + 2 * hi;
      const int pr = ks * 2 + hi;
      acc3 = wmma4(lds_b64(&myH[l15 * HP + c0]),
                   lds_b64(&sW3[pr * 32 + l15 * 2]), acc3);
    }
    if (l15 < 8) {
      const float bb = sB3[l15];
#pragma unroll
      for (int r = 0; r < 8; ++r)
        dy[(base + hi * 8 + r) * 8 + l15] = acc3[r] + bb;
    }

    // ---- butterfly-reduce divergence over the 16-lane column group ----
#pragma unroll
    for (int r = 0; r < 8; ++r) {
      float v = p[r];
      v += __shfl_xor(v, 1, 32);
      v += __shfl_xor(v, 2, 32);
      v += __shfl_xor(v, 4, 32);
      v += __shfl_xor(v, 8, 32);
      p[r] = v;
    }
    if (l15 == 0) {
#pragma unroll
      for (int r = 0; r < 8; ++r)
        negdiv[base + hi * 8 + r] = -p[r];
    }
    __syncthreads();
  }
}

extern "C" void kernel_launch(void* const* d_in, const int* in_sizes, int n_in,
                              void* d_out, int out_size, void* d_ws, size_t ws_size,
                              hipStream_t stream) {
  (void)n_in; (void)out_size; (void)d_ws; (void)ws_size;
  const float* t  = (const float*)d_in[0];
  const float* y  = (const float*)d_in[1];
  // d_in[2] = logp (unused; output does not depend on it)
  const float* W1 = (const float*)d_in[3];
  const float* b1 = (const float*)d_in[4];
  const float* W2 = (const float*)d_in[5];
  const float* b2 = (const float*)d_in[6];
  const float* W3 = (const float*)d_in[7];
  const float* b3 = (const float*)d_in[8];

  const int Btot = in_sizes[1] / 8;            // 524288
  float* out_dy = (float*)d_out;               // [B, 8]
  float* out_nd = out_dy + (size_t)Btot * 8;   // [B, 1]

  const int blocks = Btot / 256;               // 4 waves x 4 tiles x 16 samples
  odefunc_cnf_kernel<<<blocks, 128, 0, stream>>>(t, y, W1, b1, W2, b2, W3, b3,
                                                 out_dy, out_nd);
}